// PairBiasCrossAttentionPD_45208825757995
// MI455X (gfx1250) — compile-verified
//
#include <hip/hip_runtime.h>

#define LCD 1024
#define CS  768
#define NH  16
#define DH  48
#define CP  128
#define NJCHUNK 8

typedef __attribute__((ext_vector_type(2))) float v2f;
typedef __attribute__((ext_vector_type(8))) float v8f;

// A-fragment (and B-fragment from row-major [N][K] storage):
// lane l -> row = l&15, K pair = k0 + 2*(l>>4), contiguous in memory.
__device__ __forceinline__ v2f frag_rowK(const float* __restrict__ base, int ld, int k0) {
  const int l = threadIdx.x & 31;
  const float* p = base + (size_t)(l & 15) * ld + (k0 + ((l >> 4) << 1));
  return *(const v2f*)p;
}

// Same, but with non-temporal hint (stream-once data: bypass/deprioritize caches).
__device__ __forceinline__ v2f frag_rowK_nt(const float* __restrict__ base, int ld, int k0) {
  const int l = threadIdx.x & 31;
  const float* p = base + (size_t)(l & 15) * ld + (k0 + ((l >> 4) << 1));
  return __builtin_nontemporal_load((const v2f*)p);
}

// B-fragment from row-major [K][N] storage: lane l -> n = n0 + (l&15), K pair strided by ld.
__device__ __forceinline__ v2f frag_colK(const float* __restrict__ base, int ld, int n0, int k0) {
  const int l = threadIdx.x & 31;
  const int n = n0 + (l & 15);
  const int k = k0 + ((l >> 4) << 1);
  return v2f{base[(size_t)k * ld + n], base[(size_t)(k + 1) * ld + n]};
}

__device__ __forceinline__ v8f wmma_f32(v2f a, v2f b, v8f c) {
  return __builtin_amdgcn_wmma_f32_16x16x4_f32(false, a, false, b, (short)0, c, false, false);
}

// ---------------- Pass A: q,k,v projections + sigmoid gate ----------------
__global__ __launch_bounds__(256) void proj_kernel(
    const float* __restrict__ s,
    const float* __restrict__ Wq, const float* __restrict__ bq,
    const float* __restrict__ Wk, const float* __restrict__ bk,
    const float* __restrict__ Wv, const float* __restrict__ bv,
    const float* __restrict__ Wg, const float* __restrict__ bg,
    float* __restrict__ q, float* __restrict__ k,
    float* __restrict__ v, float* __restrict__ g)
{
  const int tile = blockIdx.x * 8 + (threadIdx.x >> 5);
  const int ntn = CS / 16;                 // 48 n-tiles
  const int per = (LCD / 16) * ntn;        // 3072 tiles per matrix
  const int mat = tile / per;
  const int t   = tile % per;
  const int m0  = (t / ntn) * 16;
  const int n0  = (t % ntn) * 16;
  const float* W; const float* b; float* out;
  if (mat == 0)      { W = Wq; b = bq; out = q; }
  else if (mat == 1) { W = Wk; b = bk; out = k; }
  else if (mat == 2) { W = Wv; b = bv; out = v; }
  else               { W = Wg; b = bg; out = g; }

  v8f c = {};
  for (int k0 = 0; k0 < CS; k0 += 4) {
    v2f a  = frag_rowK(s + (size_t)m0 * CS, CS, k0);
    v2f bf = frag_colK(W, CS, n0, k0);
    c = wmma_f32(a, bf, c);
  }
  const int l = threadIdx.x & 31;
  const int col = n0 + (l & 15);
  const float bias = b[col];
  #pragma unroll
  for (int r = 0; r < 8; ++r) {
    const int row = m0 + r + ((l >> 4) << 3);
    float val = c[r] + bias;
    if (mat == 3) val = __builtin_amdgcn_rcpf(1.f + __expf(-val));  // fast sigmoid
    out[(size_t)row * CS + col] = val;
  }
}

// ---------------- Pass B: logits = scale*QK^T + z@Wz + dist + prior, masked ----------------
__global__ __launch_bounds__(512) void logits_kernel(
    const float* __restrict__ q, const float* __restrict__ kmat,
    const float* __restrict__ z, const float* __restrict__ Wz,
    const float* __restrict__ dist, const float* __restrict__ prior,
    const unsigned char* __restrict__ mask,
    float* __restrict__ attn)
{
  __shared__ float lb[16 * 16 * NH];       // [i_local][j_local][head]
  const int w  = threadIdx.x >> 5;         // wave id: row in step 1, head in step 2
  const int l  = threadIdx.x & 31;
  const int bi    = blockIdx.x / NJCHUNK;
  const int chunk = blockIdx.x % NJCHUNK;
  const int i0 = bi * 16;
  const float scale = 0.14433756729740643f; // 1/sqrt(48)

  // preload Wz B-fragments (shared by all j-tiles) and q A-fragments for head w
  v2f wzf[CP / 4];
  #pragma unroll
  for (int kk = 0; kk < CP / 4; ++kk) wzf[kk] = frag_colK(Wz, NH, 0, kk * 4);
  v2f qf[DH / 4];
  #pragma unroll
  for (int kk = 0; kk < DH / 4; ++kk)
    qf[kk] = frag_rowK(q + (size_t)i0 * CS + w * DH, CS, kk * 4);

  for (int jt = 0; jt < 8; ++jt) {
    const int j0 = chunk * 128 + jt * 16;

    // --- step 1: wave w builds bias tile for row i = i0+w, all 16 heads ---
    {
      const int i = i0 + w;
      const float* zrow = z + ((size_t)i * LCD + j0) * CP;   // [16 j][128 c]
      v8f c = {};
      #pragma unroll 4
      for (int kk = 0; kk < CP / 4; ++kk)
        c = wmma_f32(frag_rowK_nt(zrow, CP, kk * 4), wzf[kk], c);   // z streamed once: NT
      const int n = l & 15;                                  // head
      #pragma unroll
      for (int r = 0; r < 8; ++r) {
        const int m = r + ((l >> 4) << 3);                   // j_local
        const size_t ij = (size_t)i * LCD + j0 + m;
        float val = c[r] + __builtin_nontemporal_load(dist + ij * NH + n)
                         + __builtin_nontemporal_load(prior + ij);
        if (!__builtin_nontemporal_load(mask + ij)) val = -3.0e38f;
        lb[(w * 16 + m) * NH + n] = val;
      }
    }
    __syncthreads();

    // --- step 2: wave h = w computes its 16x16 logits tile ---
    {
      v8f c = {};
      #pragma unroll
      for (int kk = 0; kk < DH / 4; ++kk)
        c = wmma_f32(qf[kk], frag_rowK(kmat + (size_t)j0 * CS + w * DH, CS, kk * 4), c);
      const int n = l & 15;                                  // j_local
      #pragma unroll
      for (int r = 0; r < 8; ++r) {
        const int m = r + ((l >> 4) << 3);                   // i_local
        const float val = c[r] * scale + lb[(m * 16 + n) * NH + w];
        attn[(size_t)w * LCD * LCD + (size_t)(i0 + m) * LCD + (j0 + n)] = val;
      }
    }
    __syncthreads();
  }
}

// ---------------- Pass C: row softmax over attn, in place ----------------
__global__ __launch_bounds__(256) void softmax_kernel(float* __restrict__ attn)
{
  const int row = blockIdx.x * 8 + (threadIdx.x >> 5);  // 16384 rows
  const int l = threadIdx.x & 31;
  float* p = attn + (size_t)row * LCD;
  float4 vb[8];
  float m = -3.4e38f;
  #pragma unroll
  for (int qd = 0; qd < 8; ++qd) {
    vb[qd] = ((const float4*)p)[qd * 32 + l];
    m = fmaxf(m, fmaxf(fmaxf(vb[qd].x, vb[qd].y), fmaxf(vb[qd].z, vb[qd].w)));
  }
  #pragma unroll
  for (int off = 16; off > 0; off >>= 1) m = fmaxf(m, __shfl_xor(m, off, 32));
  float sum = 0.f;
  #pragma unroll
  for (int qd = 0; qd < 8; ++qd) {
    vb[qd].x = __expf(vb[qd].x - m); vb[qd].y = __expf(vb[qd].y - m);
    vb[qd].z = __expf(vb[qd].z - m); vb[qd].w = __expf(vb[qd].w - m);
    sum += (vb[qd].x + vb[qd].y) + (vb[qd].z + vb[qd].w);
  }
  #pragma unroll
  for (int off = 16; off > 0; off >>= 1) sum += __shfl_xor(sum, off, 32);
  const float inv = 1.f / sum;
  #pragma unroll
  for (int qd = 0; qd < 8; ++qd) {
    vb[qd].x *= inv; vb[qd].y *= inv; vb[qd].z *= inv; vb[qd].w *= inv;
    ((float4*)p)[qd * 32 + l] = vb[qd];
  }
}

// ---------------- Pass D: ctx[h,i,:] = attn[h,i,:] @ v[h] ----------------
__global__ __launch_bounds__(256) void ctx_kernel(
    const float* __restrict__ attn, const float* __restrict__ v,
    float* __restrict__ ctx)
{
  const int task = blockIdx.x * 8 + (threadIdx.x >> 5);  // 16 heads * 64 i-tiles
  const int h  = task / (LCD / 16);
  const int i0 = (task % (LCD / 16)) * 16;
  const float* arow = attn + (size_t)h * LCD * LCD + (size_t)i0 * LCD;
  const float* vh = v + h * DH;
  v8f c0 = {}, c1 = {}, c2 = {};
  for (int k0 = 0; k0 < LCD; k0 += 4) {
    v2f a = frag_rowK(arow, LCD, k0);
    c0 = wmma_f32(a, frag_colK(vh, CS,  0, k0), c0);
    c1 = wmma_f32(a, frag_colK(vh, CS, 16, k0), c1);
    c2 = wmma_f32(a, frag_colK(vh, CS, 32, k0), c2);
  }
  const int l = threadIdx.x & 31;
  #pragma unroll
  for (int r = 0; r < 8; ++r) {
    const int row = i0 + r + ((l >> 4) << 3);
    float* o = ctx + (size_t)row * CS + h * DH + (l & 15);
    o[0] = c0[r]; o[16] = c1[r]; o[32] = c2[r];
  }
}

// ---------------- Pass E: y = gate * (ctx @ Wo + bo) ----------------
__global__ __launch_bounds__(256) void out_kernel(
    const float* __restrict__ ctx, const float* __restrict__ Wo,
    const float* __restrict__ bo, const float* __restrict__ gate,
    float* __restrict__ y)
{
  const int tile = blockIdx.x * 8 + (threadIdx.x >> 5);  // 3072 tiles
  const int ntn = CS / 16;
  const int m0 = (tile / ntn) * 16;
  const int n0 = (tile % ntn) * 16;
  v8f c = {};
  for (int k0 = 0; k0 < CS; k0 += 4)
    c = wmma_f32(frag_rowK(ctx + (size_t)m0 * CS, CS, k0), frag_colK(Wo, CS, n0, k0), c);
  const int l = threadIdx.x & 31;
  const int col = n0 + (l & 15);
  const float bias = bo[col];
  #pragma unroll
  for (int r = 0; r < 8; ++r) {
    const int row = m0 + r + ((l >> 4) << 3);
    y[(size_t)row * CS + col] = gate[(size_t)row * CS + col] * (c[r] + bias);
  }
}

extern "C" void kernel_launch(void* const* d_in, const int* in_sizes, int n_in,
                              void* d_out, int out_size, void* d_ws, size_t ws_size,
                              hipStream_t stream) {
  const float* s     = (const float*)d_in[0];
  const float* z     = (const float*)d_in[1];
  const unsigned char* pd = (const unsigned char*)d_in[2];
  const float* dist  = (const float*)d_in[3];
  const float* prior = (const float*)d_in[4];
  const float* Wq = (const float*)d_in[5];  const float* bq = (const float*)d_in[6];
  const float* Wk = (const float*)d_in[7];  const float* bk = (const float*)d_in[8];
  const float* Wv = (const float*)d_in[9];  const float* bv = (const float*)d_in[10];
  const float* Wz = (const float*)d_in[11];
  const float* Wo = (const float*)d_in[12]; const float* bo = (const float*)d_in[13];
  const float* Wg = (const float*)d_in[14]; const float* bg = (const float*)d_in[15];

  float* ws  = (float*)d_ws;
  float* q   = ws;
  float* k   = ws + (size_t)1 * LCD * CS;
  float* v   = ws + (size_t)2 * LCD * CS;
  float* g   = ws + (size_t)3 * LCD * CS;
  float* ctx = ws + (size_t)4 * LCD * CS;

  float* y    = (float*)d_out;
  float* attn = y + (size_t)LCD * CS;

  proj_kernel<<<1536, 256, 0, stream>>>(s, Wq, bq, Wk, bk, Wv, bv, Wg, bg, q, k, v, g);
  logits_kernel<<<(LCD / 16) * NJCHUNK, 512, 0, stream>>>(q, k, z, Wz, dist, prior, pd, attn);
  softmax_kernel<<<(NH * LCD) / 8, 256, 0, stream>>>(attn);
  ctx_kernel<<<(NH * (LCD / 16)) / 8, 256, 0, stream>>>(attn, v, ctx);
  out_kernel<<<((LCD / 16) * (CS / 16)) / 8, 256, 0, stream>>>(ctx, Wo, bo, g, y);
}